// FNO2d_6485400617142
// MI455X (gfx1250) — compile-verified
//
#include <hip/hip_runtime.h>
#include <math.h>
#include <stdint.h>

// FNO2d on MI455X (gfx1250, wave32). All GEMM-shaped work routed through
// V_WMMA_F32_16X16X4_F32 (full f32 precision, matches reference numerics).
// Spectral conv implemented as partial DFT (only 16x16 modes survive the
// truncation), i.e. GEMMs against a precomputed cos/-sin basis.
// Heavy global->LDS staging uses GLOBAL_LOAD_ASYNC_TO_LDS_B128 (ASYNCcnt).

#define BATCH  8
#define NXD    256
#define NXY    65536          // 256*256
#define WID    32
#define HID    128
#define NLAY   4
#define ROWP   260            // padded LDS row stride (bank-conflict-free)

typedef __attribute__((ext_vector_type(2))) float v2f;
typedef __attribute__((ext_vector_type(8))) float v8f;

// D = A(16x4) * B(4x16) + C  ; f32 WMMA (CDNA5)
__device__ __forceinline__ v8f wmma4(v2f a, v2f b, v8f c) {
  return __builtin_amdgcn_wmma_f32_16x16x4_f32(
      /*neg_a=*/false, a, /*neg_b=*/false, b,
      /*c_mod=*/(short)0, c, /*reuse_a=*/false, /*reuse_b=*/false);
}

__device__ __forceinline__ float gelu_exact(float v) {
  return 0.5f * v * (1.0f + erff(v * 0.70710678118654752440f));
}

// Async copy 16B from global to LDS (tracked by ASYNCcnt).
__device__ __forceinline__ void async_g2l_b128(unsigned lds_off, const float* gp) {
  asm volatile("global_load_async_to_lds_b128 %0, %1, off"
               :: "v"(lds_off), "v"(gp) : "memory");
}
__device__ __forceinline__ void wait_async0() {
  asm volatile("s_wait_asynccnt 0" ::: "memory");
}

// basis[y*32 + m]      = cos(2*pi*m*y/256)      m in [0,16)
// basis[y*32 + 16 + m] = -sin(2*pi*m*y/256)
// => columns 0..15 give DFT real part, 16..31 give imag part directly.
__global__ void k_basis(float* __restrict__ basis) {
  int y = threadIdx.x;
  #pragma unroll
  for (int m = 0; m < 16; ++m) {
    float th = 6.28318530717958647692f * (float)(m * y) * (1.0f / 256.0f);
    float s, c;
    sincosf(th, &s, &c);
    basis[y * 32 + m]      = c;
    basis[y * 32 + 16 + m] = -s;
  }
}

// Lifting: h[b,c,p] = sum_k inp[b,p,k]*W[k,c] + bias[c], inp=(bc0,bc1,bc2,x,y)
__global__ __launch_bounds__(256) void k_lift(
    const float* __restrict__ bc, const float* __restrict__ xg,
    const float* __restrict__ yg, const float* __restrict__ w,
    const float* __restrict__ bias, float* __restrict__ h) {
  int g = blockIdx.x * 256 + threadIdx.x;   // g = b*NXY + p
  int b = g >> 16;
  float i0 = bc[b * 3 + 0], i1 = bc[b * 3 + 1], i2 = bc[b * 3 + 2];
  float i3 = xg[g], i4 = yg[g];
  int p = g & (NXY - 1);
  #pragma unroll 4
  for (int c = 0; c < WID; ++c) {
    float acc = bias[c] + i0 * w[c] + i1 * w[32 + c] + i2 * w[64 + c]
              + i3 * w[96 + c] + i4 * w[128 + c];
    h[(size_t)(b * WID + c) * NXY + p] = acc;   // coalesced along p per c
  }
}

// Forward DFT along y as WMMA GEMM: per plane (b,c):
//   A[x, 0:16]=Re, A[x,16:32]=Im of sum_y h[x,y]*e^{-2pi i m2 y/256}
// M=256 (x), K=256 (y), N=32.  Block stages a contiguous 64-row slab of H
// into LDS via async b128 copies (padded rows: ds reads conflict-free),
// then 8 waves compute 4 row-tiles x 2 col-tiles.
__global__ __launch_bounds__(256) void k_fwd_y(
    const float* __restrict__ h, const float* __restrict__ basis,
    float* __restrict__ A) {
  __shared__ float sh[64 * ROWP];             // 65 KB slab, padded rows
  int plane = blockIdx.x >> 2;                // b*32+c
  int sub   = blockIdx.x & 3;                 // which 64-row slab
  int t     = threadIdx.x;
  const float* slab = h + (size_t)plane * NXY + (size_t)(sub * 64) * NXD;
  unsigned lds_base = (unsigned)(uintptr_t)sh;
  #pragma unroll
  for (int r = 0; r < 16; ++r) {              // 64KB contiguous -> LDS (padded)
    int g = r * 1024 + t * 4;                 // float index in slab
    int row = g >> 8, y = g & 255;
    async_g2l_b128(lds_base + (unsigned)(row * ROWP + y) * 4u, slab + g);
  }
  wait_async0();
  __syncthreads();

  int wave = t >> 5, lane = t & 31;
  int l16 = lane & 15, lh = lane >> 4;
  int trl = wave >> 1, tcol = wave & 1;       // local row-tile 0..3, col-tile 0..1
  const float* srow = sh + (trl * 16 + l16) * ROWP;
  int cb = tcol * 16 + l16;
  v8f acc = {};
  #pragma unroll 4
  for (int s = 0; s < 64; ++s) {
    int kb = s * 4 + lh * 2;
    v2f a = *(const v2f*)(srow + kb);         // LDS b64, bank-conflict-free
    v2f bb;
    bb[0] = basis[kb * 32 + cb];
    bb[1] = basis[(kb + 1) * 32 + cb];
    acc = wmma4(a, bb, acc);
  }
  float* Ap = A + (size_t)plane * NXD * 32;
  int row0 = sub * 64 + trl * 16;
  #pragma unroll
  for (int v = 0; v < 8; ++v)
    Ap[(row0 + v + 8 * lh) * 32 + cb] = acc[v];
}

// Forward DFT along x (complex, 16 modes): F[b,c,m1,m2] = (1/256)*sum_x A[x]*e^{-i th}
__global__ __launch_bounds__(256) void k_fwd_x(
    const float* __restrict__ A, const float* __restrict__ basis,
    float* __restrict__ Fre, float* __restrict__ Fim) {
  int g = blockIdx.x * 256 + threadIdx.x;     // plane*256 + m1*16 + m2
  int m2 = g & 15, m1 = (g >> 4) & 15;
  int plane = g >> 8;
  const float* Ap = A + (size_t)plane * NXD * 32;
  float fr = 0.f, fi = 0.f;
  for (int x = 0; x < NXD; ++x) {
    float ar = Ap[x * 32 + m2];
    float ai = Ap[x * 32 + 16 + m2];
    float cc = basis[x * 32 + m1];
    float ss = basis[x * 32 + 16 + m1];       // = -sin -> e^{-i th} = cc + i*ss
    fr += ar * cc - ai * ss;
    fi += ar * ss + ai * cc;
  }
  Fre[g] = fr * (1.0f / 256.0f);              // full rfft2 ortho scale
  Fim[g] = fi * (1.0f / 256.0f);
}

// Complex mode mixing: T[b,o,m] = sum_i F[b,i,m] * (wr+i wi)[i,o,m]
__global__ __launch_bounds__(256) void k_modemix(
    const float* __restrict__ Fre, const float* __restrict__ Fim,
    const float* __restrict__ wr, const float* __restrict__ wi,
    float* __restrict__ Tre, float* __restrict__ Tim) {
  int g = blockIdx.x * 256 + threadIdx.x;     // (b*32+o)*256 + mm
  int mm = g & 255;
  int o = (g >> 8) & 31;
  int b = g >> 13;
  float tr = 0.f, ti = 0.f;
  for (int i = 0; i < WID; ++i) {
    float xr = Fre[(b * WID + i) * 256 + mm];
    float xi = Fim[(b * WID + i) * 256 + mm];
    float ur = wr[(size_t)(i * WID + o) * 256 + mm];
    float ui = wi[(size_t)(i * WID + o) * 256 + mm];
    tr += xr * ur - xi * ui;
    ti += xr * ui + xi * ur;
  }
  Tre[g] = tr;
  Tim[g] = ti;
}

// Inverse DFT along x (16 modes, full complex ifft, ortho 1/16):
// G[b,o,x,m2] = (1/16) * sum_m1 T[m1,m2] * e^{+2pi i m1 x/256}
__global__ __launch_bounds__(256) void k_inv_x(
    const float* __restrict__ Tre, const float* __restrict__ Tim,
    const float* __restrict__ basis, float* __restrict__ G) {
  int g = blockIdx.x * 256 + threadIdx.x;     // plane*4096 + x*16 + m2
  int m2 = g & 15;
  int x = (g >> 4) & 255;
  int plane = g >> 12;
  const float* tr = Tre + plane * 256;
  const float* ti = Tim + plane * 256;
  float gr = 0.f, gi = 0.f;
  #pragma unroll
  for (int m1 = 0; m1 < 16; ++m1) {
    float trv = tr[m1 * 16 + m2], tiv = ti[m1 * 16 + m2];
    float cc = basis[x * 32 + m1];
    float ns = basis[x * 32 + 16 + m1];       // -sin
    gr += trv * cc + tiv * ns;                // tr*cos - ti*sin
    gi += tiv * cc - trv * ns;                // tr*sin + ti*cos
  }
  float* Gp = G + (size_t)plane * NXD * 32;
  Gp[x * 32 + m2]      = gr * (1.0f / 16.0f);
  Gp[x * 32 + 16 + m2] = gi * (1.0f / 16.0f);
}

// Pointwise conv as WMMA GEMM: per batch tmp(32 x 65536) = pw_w(32x32) * h(32x65536)
__global__ __launch_bounds__(256) void k_pw(
    const float* __restrict__ h, const float* __restrict__ pww,
    float* __restrict__ tmp) {
  int T = blockIdx.x * 8 + (threadIdx.x >> 5);
  int lane = threadIdx.x & 31;
  int l16 = lane & 15, lh = lane >> 4;
  int b  = T >> 13;                            // 8192 tiles per batch
  int r  = T & 8191;
  int ct = r & 4095, rt = r >> 12;
  int p0 = ct * 16;
  const float* Hb = h + (size_t)b * WID * NXY;
  int arow = rt * 16 + l16;
  v8f acc = {};
  #pragma unroll
  for (int s = 0; s < 8; ++s) {
    int kb = s * 4 + lh * 2;
    v2f a = *(const v2f*)(pww + arow * 32 + kb);   // pw_w[o][i], 8B aligned
    v2f bb;
    bb[0] = Hb[(size_t)kb * NXY + p0 + l16];       // coalesced 64B per K-row
    bb[1] = Hb[(size_t)(kb + 1) * NXY + p0 + l16];
    acc = wmma4(a, bb, acc);
  }
  float* Ob = tmp + (size_t)b * WID * NXY;
  #pragma unroll
  for (int v = 0; v < 8; ++v)
    Ob[(size_t)(rt * 16 + v + 8 * lh) * NXY + p0 + l16] = acc[v];
}

// Fused: inverse-rfft along y (16 modes, ortho 1/16) + pw + bias + GELU + residual
__global__ __launch_bounds__(256) void k_invy_gelu_res(
    const float* __restrict__ G, const float* __restrict__ tmp,
    const float* __restrict__ pwb, const float* __restrict__ basis,
    float* __restrict__ h) {
  int g = blockIdx.x * 256 + threadIdx.x;     // plane*NXY + p
  int p = g & (NXY - 1);
  int plane = g >> 16;
  int o = plane & 31;
  int x = p >> 8, y = p & 255;
  const float* Gp = G + (size_t)plane * NXD * 32 + x * 32;
  float sc = Gp[0];                           // m2=0 term (Re only)
  #pragma unroll
  for (int m2 = 1; m2 < 16; ++m2) {
    float cc = basis[y * 32 + m2];
    float ns = basis[y * 32 + 16 + m2];       // -sin
    sc += 2.0f * (Gp[m2] * cc + Gp[16 + m2] * ns);  // 2*Re(X*e^{+i th})
  }
  float u = sc * (1.0f / 16.0f) + tmp[g] + pwb[o];
  h[g] += gelu_exact(u);
}

// Fully fused 3-head decoder. One block per 128-point tile:
// async LDS h-tile -> WMMA W1^T*H -> GELU -> WMMA W2^T*Z1 -> GELU*w3 GEMV
// fused in the epilogue (cross-half reduce via shfl_xor 16).
__global__ __launch_bounds__(256) void k_decoder(
    const float* __restrict__ h,
    const float* __restrict__ w1, const float* __restrict__ b1,
    const float* __restrict__ w2, const float* __restrict__ b2,
    const float* __restrict__ w3, const float* __restrict__ b3,
    float* __restrict__ out) {
  __shared__ float sh_h[WID * 128];
  __shared__ float sh_z[WID * 128];
  int blk = blockIdx.x;
  int b = blk >> 9;                            // 512 tiles per batch
  int p0 = (blk & 511) << 7;
  int t = threadIdx.x;
  const float* Hb = h + (size_t)b * WID * NXY + p0;
  unsigned lds_base = (unsigned)(uintptr_t)sh_h;
  #pragma unroll
  for (int r = 0; r < 4; ++r) {                // 16KB tile via async b128
    int idx = (t + r * 256) * 4;               // = c*128 + p, p mult of 4
    async_g2l_b128(lds_base + (unsigned)idx * 4u,
                   Hb + (size_t)(idx >> 7) * NXY + (idx & 127));
  }
  wait_async0();
  __syncthreads();
  int wave = t >> 5, lane = t & 31;
  int l16 = lane & 15, lh = (lane >> 4) & 1;
  for (int d = 0; d < 3; ++d) {
    const float* W1 = w1 + d * WID * WID;      // [i][o] -> A^T[o][i]
    #pragma unroll
    for (int q = 0; q < 2; ++q) {
      int tt = q * 8 + wave;                   // 16 tiles: 2 row x 8 col
      int rowb = (tt >> 3) * 16, colb = (tt & 7) * 16;
      v8f acc = {};
      #pragma unroll
      for (int s = 0; s < 8; ++s) {
        int kb = s * 4 + lh * 2;
        v2f a, bb;
        a[0]  = W1[kb * WID + rowb + l16];
        a[1]  = W1[(kb + 1) * WID + rowb + l16];
        bb[0] = sh_h[kb * 128 + colb + l16];
        bb[1] = sh_h[(kb + 1) * 128 + colb + l16];
        acc = wmma4(a, bb, acc);
      }
      #pragma unroll
      for (int v = 0; v < 8; ++v) {
        int row = rowb + v + 8 * lh;
        sh_z[row * 128 + colb + l16] = gelu_exact(acc[v] + b1[d * WID + row]);
      }
    }
    __syncthreads();
    const float* W2 = w2 + d * WID * HID;      // [o][j] -> A^T[j][o]
    float pg = 0.f;
    int colb = wave * 16;                      // each wave owns one col-tile
    #pragma unroll
    for (int q = 0; q < 8; ++q) {              // 8 row-tiles of Z2 (128 rows)
      int rowb = q * 16;
      v8f acc = {};
      #pragma unroll
      for (int s = 0; s < 8; ++s) {
        int kb = s * 4 + lh * 2;
        v2f a, bb;
        a[0]  = W2[kb * HID + rowb + l16];
        a[1]  = W2[(kb + 1) * HID + rowb + l16];
        bb[0] = sh_z[kb * 128 + colb + l16];
        bb[1] = sh_z[(kb + 1) * 128 + colb + l16];
        acc = wmma4(a, bb, acc);
      }
      #pragma unroll
      for (int v = 0; v < 8; ++v) {            // fused GEMV epilogue
        int j = rowb + v + 8 * lh;
        pg += gelu_exact(acc[v] + b2[d * HID + j]) * w3[d * HID + j];
      }
    }
    float tot = pg + __shfl_xor(pg, 16, 32);   // combine half-wave row splits
    if (lane < 16) {
      int p = p0 + colb + l16;
      out[(size_t)(b * NXY + p) * 3 + d] = tot + b3[d];
    }
    __syncthreads();
  }
}

extern "C" void kernel_launch(void* const* d_in, const int* in_sizes, int n_in,
                              void* d_out, int out_size, void* d_ws, size_t ws_size,
                              hipStream_t stream) {
  (void)in_sizes; (void)n_in; (void)out_size; (void)ws_size;
  const float* bc  = (const float*)d_in[0];
  const float* xg  = (const float*)d_in[1];
  const float* yg  = (const float*)d_in[2];
  const float* m1w = (const float*)d_in[3];
  const float* m1b = (const float*)d_in[4];
  const float* swr = (const float*)d_in[5];
  const float* swi = (const float*)d_in[6];
  const float* pww = (const float*)d_in[7];
  const float* pwb = (const float*)d_in[8];
  const float* dw1 = (const float*)d_in[9];
  const float* db1 = (const float*)d_in[10];
  const float* dw2 = (const float*)d_in[11];
  const float* db2 = (const float*)d_in[12];
  const float* dw3 = (const float*)d_in[13];
  const float* db3 = (const float*)d_in[14];
  float* out = (float*)d_out;

  float* ws = (float*)d_ws;
  float* basis = ws;  ws += 8192;                       // 256x32 DFT basis
  float* h     = ws;  ws += (size_t)BATCH * WID * NXY;  // 64 MB state
  float* tmp   = ws;  ws += (size_t)BATCH * WID * NXY;  // 64 MB pw output
  float* AG    = ws;  ws += (size_t)BATCH * WID * NXD * 32; // fwd-y A / inv-x G (aliased)
  float* Fre   = ws;  ws += 65536;
  float* Fim   = ws;  ws += 65536;
  float* Tre   = ws;  ws += 65536;
  float* Tim   = ws;  ws += 65536;

  k_basis<<<1, 256, 0, stream>>>(basis);
  k_lift<<<BATCH * NXY / 256, 256, 0, stream>>>(bc, xg, yg, m1w, m1b, h);
  for (int l = 0; l < NLAY; ++l) {
    k_fwd_y<<<1024, 256, 0, stream>>>(h, basis, AG);
    k_fwd_x<<<256, 256, 0, stream>>>(AG, basis, Fre, Fim);
    k_modemix<<<256, 256, 0, stream>>>(Fre, Fim,
                                       swr + (size_t)l * WID * WID * 256,
                                       swi + (size_t)l * WID * WID * 256,
                                       Tre, Tim);
    k_inv_x<<<4096, 256, 0, stream>>>(Tre, Tim, basis, AG);
    k_pw<<<8192, 256, 0, stream>>>(h, pww + l * WID * WID, tmp);
    k_invy_gelu_res<<<65536, 256, 0, stream>>>(AG, tmp, pwb + l * WID, basis, h);
  }
  k_decoder<<<BATCH * 512, 256, 0, stream>>>(h, dw1, db1, dw2, db2, dw3, db3, out);
}